// QuantizationLayer_3770981286078
// MI455X (gfx1250) — compile-verified
//
#include <hip/hip_runtime.h>
#include <hip/hip_bf16.h>

// ---------------------------------------------------------------------------
// Problem constants (from reference): B=16, T=1024, D=512, G=2, V=320, DV=128
// N = B*T = 16384 tokens, E = G*V = 640 logit columns, OUT = 256
// ---------------------------------------------------------------------------
#define NROWS 16384
#define DDIM  512
#define EDIM  640
#define VDIM  320
#define DVDIM 128

typedef float v2f __attribute__((ext_vector_type(2)));
typedef float v8f __attribute__((ext_vector_type(8)));

// ---------------------------------------------------------------------------
// Kernel 1: logits = x @ W^T + b   via V_WMMA_F32_16X16X4_F32
//
// One wave -> 16(rows) x 64(cols) output tile = 4 accumulators of 16x16.
// A fragment (16x4 of x) is loaded once per K-step and reused for 4 WMMAs.
//
// ISA A/B fragment layout for 32-bit 16x4 / 4x16 (wave32):
//   lane m = lane&15 selects row (A: M) / col (B: N)
//   half h = lane>>4 selects K pair: v0 = K(2h), v1 = K(2h+1)
// Both x and W are row-major over D, and B[k][e] = W[e][k], so A and B
// fragment loads use the *same* addressing pattern on x / W respectively.
// ---------------------------------------------------------------------------
__global__ __launch_bounds__(256)
void vq_gemm_logits_kernel(const float* __restrict__ x,
                           const float* __restrict__ W,
                           const float* __restrict__ bias,
                           float* __restrict__ logits) {
  const int waveInBlock = threadIdx.x >> 5;
  const int lane        = threadIdx.x & 31;
  const int waveId      = blockIdx.x * 8 + waveInBlock;   // 8 waves / block

  // 1024 row tiles (16 rows each) x 10 col tiles (64 cols each)
  const int etile = waveId % 10;
  const int ntile = waveId / 10;

  const int m = lane & 15;   // row within A tile / col within B tile
  const int h = lane >> 4;   // K-pair selector

  const float* aptr = x + (size_t)(ntile * 16 + m) * DDIM + 2 * h;
  const float* bptr = W + (size_t)(etile * 64 + m) * DDIM + 2 * h;

  v8f acc0 = {}, acc1 = {}, acc2 = {}, acc3 = {};

  #pragma unroll 4
  for (int k = 0; k < DDIM; k += 4) {
    v2f a  = *(const v2f*)(aptr + k);
    v2f b0 = *(const v2f*)(bptr + k);
    v2f b1 = *(const v2f*)(bptr + 16 * DDIM + k);
    v2f b2 = *(const v2f*)(bptr + 32 * DDIM + k);
    v2f b3 = *(const v2f*)(bptr + 48 * DDIM + k);
    acc0 = __builtin_amdgcn_wmma_f32_16x16x4_f32(false, a, false, b0,
                                                 (short)0, acc0, false, false);
    acc1 = __builtin_amdgcn_wmma_f32_16x16x4_f32(false, a, false, b1,
                                                 (short)0, acc1, false, false);
    acc2 = __builtin_amdgcn_wmma_f32_16x16x4_f32(false, a, false, b2,
                                                 (short)0, acc2, false, false);
    acc3 = __builtin_amdgcn_wmma_f32_16x16x4_f32(false, a, false, b3,
                                                 (short)0, acc3, false, false);
  }

  // C/D layout: acc[r] -> row (r + 8*h), col = m (within the 16-wide tile)
  const int e0   = etile * 64;
  const int rbase = ntile * 16 + 8 * h;
  v8f accs[4] = {acc0, acc1, acc2, acc3};
  #pragma unroll
  for (int t = 0; t < 4; ++t) {
    const int col = e0 + t * 16 + m;
    const float bv = bias[col];
    #pragma unroll
    for (int r = 0; r < 8; ++r) {
      logits[(size_t)(rbase + r) * EDIM + col] = accs[t][r] + bv;
    }
  }
}

// ---------------------------------------------------------------------------
// Kernel 2: per (n,g) row of V=320 logits:
//   idx = argmax(logits + gumbel(u)),  gumbel = -log(-log(u+eps)+eps)
//   q[n, g*128:(g+1)*128] = codebook[g*V + idx, :]
// One wave32 per row; butterfly arg-max with first-index tie-break
// (matches jnp.argmax semantics).
// ---------------------------------------------------------------------------
__global__ __launch_bounds__(256)
void vq_argmax_gather_kernel(const float* __restrict__ logits,
                             const float* __restrict__ u,
                             const float* __restrict__ codebook,
                             float* __restrict__ q) {
  const int row  = blockIdx.x * 8 + (threadIdx.x >> 5);  // 0 .. N*G-1
  const int lane = threadIdx.x & 31;

  const float* lrow = logits + (size_t)row * VDIM;
  const float* urow = u      + (size_t)row * VDIM;

  float best = -__builtin_inff();
  int   bidx = 0x7fffffff;
  #pragma unroll
  for (int i = lane; i < VDIM; i += 32) {
    const float uv  = urow[i];
    const float gum = -__logf(-__logf(uv + 1e-10f) + 1e-10f);
    const float z   = lrow[i] + gum;
    if (z > best) { best = z; bidx = i; }   // strict >: keeps first max
  }
  #pragma unroll
  for (int off = 16; off >= 1; off >>= 1) {
    const float ob = __shfl_xor(best, off, 32);
    const int   oi = __shfl_xor(bidx, off, 32);
    if (ob > best || (ob == best && oi < bidx)) { best = ob; bidx = oi; }
  }

  // gather one 128-float codebook row: 32 lanes x float4 = 512B
  const int g = row & 1;          // row = n*G + g  (G == 2)
  const int n = row >> 1;
  const float4* src = (const float4*)(codebook +
                        ((size_t)(g * VDIM + bidx)) * DVDIM);
  float4* dst = (float4*)(q + (size_t)n * 256 + g * DVDIM);
  dst[lane] = src[lane];
}

// ---------------------------------------------------------------------------
// Launch: inputs are {x, u, W, b, codebook, temp} (temp unused: argmax is
// invariant to the positive temperature scale). Output tuple (q, logits) is
// concatenated flat in d_out: q first (16384*256), then logits (16384*640).
// ---------------------------------------------------------------------------
extern "C" void kernel_launch(void* const* d_in, const int* in_sizes, int n_in,
                              void* d_out, int out_size, void* d_ws, size_t ws_size,
                              hipStream_t stream) {
  const float* x        = (const float*)d_in[0];
  const float* u        = (const float*)d_in[1];
  const float* W        = (const float*)d_in[2];
  const float* b        = (const float*)d_in[3];
  const float* codebook = (const float*)d_in[4];

  float* out_q      = (float*)d_out;
  float* out_logits = out_q + (size_t)NROWS * 256;

  // GEMM: 1024 row-tiles * 10 col-tiles = 10240 waves, 8 waves/block
  vq_gemm_logits_kernel<<<dim3(10240 / 8), dim3(256), 0, stream>>>(
      x, W, b, out_logits);

  // argmax + gather: 32768 rows, 8 waves/block
  vq_argmax_gather_kernel<<<dim3(32768 / 8), dim3(256), 0, stream>>>(
      out_logits, u, codebook, out_q);
}